// EvidenceLevelAttention_85718957293806
// MI455X (gfx1250) — compile-verified
//
#include <hip/hip_runtime.h>

typedef float v2f __attribute__((ext_vector_type(2)));
typedef float v8f __attribute__((ext_vector_type(8)));

#define BB   8
#define NN   256
#define HH   300
#define MTOT (BB * NN)        // 2048 rows when (b,n) flattened
#define NTH  19               // ceil(300/16) column tiles
#define LDP  17               // padded LDS row stride (coprime with 64 banks)

__device__ __forceinline__ v8f wmma_f32(v2f a, v2f b, v8f c) {
  // D = A(16x4) * B(4x16) + C, full fp32 via V_WMMA_F32_16X16X4_F32
  return __builtin_amdgcn_wmma_f32_16x16x4_f32(
      /*neg_a=*/false, a, /*neg_b=*/false, b,
      /*c_mod=*/(short)0, c, /*reuse_a=*/false, /*reuse_b=*/false);
}

// ---------------------------------------------------------------------------
// K1: ai = h @ W0a^T ; ajb = h @ W0b^T + b0  (2048x300 each)
// Block = one 16-wide n-tile x 8 m-tiles (8 waves). W0a/W0b panels staged to
// LDS once (coalesced, zero-padded, stride-17 to avoid bank conflicts), so
// the K-loop is pure b64 A-load + paired ds_load + 2 WMMA with no guards.
// A-stream uses an advancing base pointer so loads get immediate offsets.
// ajb stored transposed [k][m] so K2 reads it coalesced across j-lanes.
// ---------------------------------------------------------------------------
__global__ void __launch_bounds__(256) k1_proj(
    const float* __restrict__ hp, const float* __restrict__ W0,
    const float* __restrict__ b0, float* __restrict__ AI,
    float* __restrict__ AJBT) {
  __shared__ float sBa[HH * LDP];
  __shared__ float sBb[HH * LDP];

  const int tid  = threadIdx.x;
  const int lane = tid & 31;
  const int wv   = tid >> 5;
  const int nT   = blockIdx.x % NTH;
  const int mG   = blockIdx.x / NTH;          // 0..15, 8 m-tiles each
  const int nBase = nT * 16;

  // Stage B panels: sB[k][n] = W0[nBase+n][k] (and +HH for W0b), zero-padded.
  for (int t = tid; t < 16 * HH; t += 256) {
    const int n  = t / HH;
    const int k  = t - n * HH;                // k contiguous per thread group
    const int gn = nBase + n;
    const int ga = (gn < HH ? gn : HH - 1) * (2 * HH);
    float va = W0[ga + k];
    float vb = W0[ga + HH + k];
    if (gn >= HH) { va = 0.0f; vb = 0.0f; }
    sBa[k * LDP + n] = va;
    sBb[k * LDP + n] = vb;
  }
  __syncthreads();

  const int lo  = lane & 15;
  const int hi  = lane >> 4;
  const int mT  = mG * 8 + wv;
  const int row = mT * 16 + lo;               // A row (flattened b*N+n)
  const int col = nBase + lo;                 // output column k
  const bool cok = col < HH;

  v8f cai = {};
  float bv = b0[col < HH ? col : HH - 1];
  if (!cok) bv = 0.0f;
  v8f cajb;
#pragma unroll
  for (int r = 0; r < 8; ++r) cajb[r] = bv;

  const float* pa = hp + row * HH + hi * 2;   // advancing A base (imm offsets)
  int ib = hi * LDP + lo;                     // advancing LDS index

#pragma unroll 5
  for (int k0 = 0; k0 < HH; k0 += 4) {
    v2f a;
    a.x = pa[0];
    a.y = pa[1];
    v2f ba, bb;
    ba.x = sBa[ib];
    ba.y = sBa[ib + 2 * LDP];
    bb.x = sBb[ib];
    bb.y = sBb[ib + 2 * LDP];
    cai  = wmma_f32(a, ba, cai);
    cajb = wmma_f32(a, bb, cajb);
    pa += 4;
    ib += 4 * LDP;
  }

  if (cok) {
#pragma unroll
    for (int r = 0; r < 8; ++r) {
      const int m = mT * 16 + hi * 8 + r;
      AI[m * HH + col]     = cai[r];
      AJBT[col * MTOT + m] = cajb[r];
    }
  }
}

// ---------------------------------------------------------------------------
// K2: p[b,i,j] = sum_k relu(ai[b,i,k] + ajb[b,j,k]) * w1[k] + b1 ; softmax_j
// relu sits inside the contraction -> pure VALU (add,max,fma). Each block
// owns 8 i-rows of one batch; the ajb_T stream is read once per block and
// reused across all 8 rows. Wave-shuffle reductions + 8-slot LDS softmax.
// ---------------------------------------------------------------------------
__device__ __forceinline__ float waveMax(float v) {
#pragma unroll
  for (int m = 16; m >= 1; m >>= 1) v = fmaxf(v, __shfl_xor(v, m, 32));
  return v;
}
__device__ __forceinline__ float waveSum(float v) {
#pragma unroll
  for (int m = 16; m >= 1; m >>= 1) v += __shfl_xor(v, m, 32);
  return v;
}

__global__ void __launch_bounds__(256) k2_p_softmax(
    const float* __restrict__ AI, const float* __restrict__ AJBT,
    const float* __restrict__ W1, const float* __restrict__ b1,
    float* __restrict__ Aout) {
  __shared__ float sAi[8 * HH];
  __shared__ float sW1[HH];
  __shared__ float sM[8];
  __shared__ float sS[8];

  const int tid = threadIdx.x;            // == j
  const int wv  = tid >> 5;
  const int b   = blockIdx.x >> 5;        // 32 blocks per batch (256 i / 8)
  const int i0  = (blockIdx.x & 31) * 8;

  for (int t = tid; t < HH; t += NN) sW1[t] = W1[t];
  for (int t = tid; t < 8 * HH; t += NN) {
    const int r = t / HH, k = t - r * HH;
    sAi[t] = AI[(b * NN + i0 + r) * HH + k];
  }
  __syncthreads();

  float acc[8];
#pragma unroll
  for (int r = 0; r < 8; ++r) acc[r] = 0.0f;

  const float* pj = AJBT + b * NN + tid;  // advancing coalesced stream
#pragma unroll 4
  for (int k = 0; k < HH; ++k) {
    const float aj = pj[0];
    const float w  = sW1[k];
#pragma unroll
    for (int r = 0; r < 8; ++r) {
      const float v = sAi[r * HH + k] + aj;
      acc[r] = fmaf(fmaxf(v, 0.0f), w, acc[r]);
    }
    pj += MTOT;
  }

  const float b1v = b1[0];
#pragma unroll 1
  for (int r = 0; r < 8; ++r) {
    const float p  = acc[r] + b1v;
    const float wm = waveMax(p);
    if ((tid & 31) == 0) sM[wv] = wm;
    __syncthreads();
    float mx = sM[0];
#pragma unroll
    for (int w = 1; w < 8; ++w) mx = fmaxf(mx, sM[w]);
    const float e  = __expf(p - mx);
    const float se = waveSum(e);
    if ((tid & 31) == 0) sS[wv] = se;
    __syncthreads();
    float tot = sS[0];
#pragma unroll
    for (int w = 1; w < 8; ++w) tot += sS[w];
    Aout[(b * NN + i0 + r) * NN + tid] = e * (1.0f / tot);
  }
}

// ---------------------------------------------------------------------------
// K3: out[b] = a[b] (256x256) @ h_prev[b] (256x300). Block = one n-tile of
// one batch x 8 m-tiles; h_prev panel (256x16) staged to LDS (zero-padded,
// stride-17), K-loop is guard-free b64 A-load + paired ds_load + 1 WMMA.
// ---------------------------------------------------------------------------
__global__ void __launch_bounds__(256) k3_av(
    const float* __restrict__ Aout, const float* __restrict__ hp,
    float* __restrict__ out) {
  __shared__ float sB[NN * LDP];

  const int tid  = threadIdx.x;
  const int lane = tid & 31;
  const int wv   = tid >> 5;
  const int nT   = blockIdx.x % NTH;
  const int rest = blockIdx.x / NTH;      // 0..15
  const int mG   = rest & 1;
  const int b    = rest >> 1;
  const int nBase = nT * 16;

  // Stage sB[k][n] = hp[b][k][nBase+n], zero-padded past HH.
  for (int t = tid; t < NN * 16; t += 256) {
    const int k  = t >> 4;
    const int n  = t & 15;
    const int gn = nBase + n;
    float v = hp[(b * NN + k) * HH + (gn < HH ? gn : HH - 1)];
    if (gn >= HH) v = 0.0f;
    sB[k * LDP + n] = v;
  }
  __syncthreads();

  const int lo  = lane & 15;
  const int hi  = lane >> 4;
  const int mT  = mG * 8 + wv;
  const int row = b * NN + mT * 16 + lo;  // row into a (flattened b,i)
  const int col = nBase + lo;             // h column

  v8f c = {};
  const float* pa = Aout + row * NN + hi * 2;
  int ib = hi * LDP + lo;

#pragma unroll 8
  for (int k0 = 0; k0 < NN; k0 += 4) {
    v2f a;
    a.x = pa[0];
    a.y = pa[1];
    v2f bf;
    bf.x = sB[ib];
    bf.y = sB[ib + 2 * LDP];
    c = wmma_f32(a, bf, c);
    pa += 4;
    ib += 4 * LDP;
  }

  if (col < HH) {
#pragma unroll
    for (int r = 0; r < 8; ++r)
      out[(b * NN + mT * 16 + hi * 8 + r) * HH + col] = c[r];
  }
}

// ---------------------------------------------------------------------------
extern "C" void kernel_launch(void* const* d_in, const int* in_sizes, int n_in,
                              void* d_out, int out_size, void* d_ws, size_t ws_size,
                              hipStream_t stream) {
  const float* hp = (const float*)d_in[0];   // (8,256,300)
  const float* W0 = (const float*)d_in[1];   // (300,600)
  const float* b0 = (const float*)d_in[2];   // (300,)
  const float* W1 = (const float*)d_in[3];   // (1,300)
  const float* b1 = (const float*)d_in[4];   // (1,)
  float* out = (float*)d_out;                // (8,256,300)

  float* AI   = (float*)d_ws;                // 2048*300
  float* AJBT = AI + (size_t)MTOT * HH;      // 300*2048 (transposed)
  float* Aout = AJBT + (size_t)HH * MTOT;    // 8*256*256

  // K1: 16 m-groups * 19 n-tiles = 304 blocks of 8 waves
  k1_proj<<<304, 256, 0, stream>>>(hp, W0, b0, AI, AJBT);
  // K2: 2048 rows / 8 rows-per-block = 256 blocks of 256 threads
  k2_p_softmax<<<256, 256, 0, stream>>>(AI, AJBT, W1, b1, Aout);
  // K3: 8 batches * 2 m-groups * 19 n-tiles = 304 blocks of 8 waves
  k3_av<<<304, 256, 0, stream>>>(Aout, hp, out);
}